// LQR_Solver_8091718386398
// MI455X (gfx1250) — compile-verified
//
#include <hip/hip_runtime.h>

// LQR KKT solve via Riccati recursion, single workgroup, fp32 WMMA tiles.
// n_state=32, n_input=16, n_all=48, T=64 (fixed by the reference problem).

typedef __attribute__((ext_vector_type(2))) float v2f;
typedef __attribute__((ext_vector_type(8))) float v8f;

#define NS 32
#define NI 16
#define NA 48
#define TT 64
#define LD1 49   // padded stride for 48-wide LDS arrays (odd -> conflict-free strided reads)
#define LD2 33   // padded stride for 32-wide LDS arrays

// ---- 16x16 f32 tile helpers using V_WMMA_F32_16X16X4_F32 ----
// A operand: 16x4 (MxK): lanes 0-15 -> M=lane, K={k0,k0+1}; lanes 16-31 -> M=lane-16, K={k0+2,k0+3}
// B operand: 4x16 (KxN): same K striping, N = lane&15
// C/D: VGPR r, lane<16 -> (M=r, N=lane); lane>=16 -> (M=r+8, N=lane-16)

__device__ __forceinline__ v8f wmma_k(const float* __restrict__ A, int a_rs, int a_cs,
                                      const float* __restrict__ B, int b_rs, int b_cs,
                                      int K, v8f acc) {
  const int lane = threadIdx.x & 31;
  const int half = lane >> 4;
  const int l    = lane & 15;
  for (int k0 = 0; k0 < K; k0 += 4) {
    const int ka = k0 + 2 * half;
    v2f a, b;
    a.x = A[l * a_rs + (ka + 0) * a_cs];
    a.y = A[l * a_rs + (ka + 1) * a_cs];
    b.x = B[(ka + 0) * b_rs + l * b_cs];
    b.y = B[(ka + 1) * b_rs + l * b_cs];
    acc = __builtin_amdgcn_wmma_f32_16x16x4_f32(false, a, false, b, (short)0, acc,
                                                false, false);
  }
  return acc;
}

__device__ __forceinline__ v8f load_tile(const float* base, int ld) {
  const int lane = threadIdx.x & 31, half = lane >> 4, l = lane & 15;
  v8f c;
#pragma unroll
  for (int r = 0; r < 8; ++r) c[r] = base[(r + 8 * half) * ld + l];
  return c;
}

__device__ __forceinline__ void store_tile(float* base, int ld, v8f c) {
  const int lane = threadIdx.x & 31, half = lane >> 4, l = lane & 15;
#pragma unroll
  for (int r = 0; r < 8; ++r) base[(r + 8 * half) * ld + l] = c[r];
}

__global__ __launch_bounds__(256) void lqr_riccati_kernel(
    const float* __restrict__ Ag, const float* __restrict__ Bg,
    const float* __restrict__ Cg, const float* __restrict__ x0g,
    float* __restrict__ out, float* __restrict__ ws) {
  __shared__ float sAB[NS][LD1];   // [A | B], 32 x 48
  __shared__ float sP[NS][LD2];    // cost-to-go P_{t+1}
  __shared__ float sPAB[NS][LD1];  // P @ [A B]
  __shared__ float sG[NA][LD1];    // C_t + ABt P AB
  __shared__ float sCt[NA][LD1];   // staged C_t
  __shared__ float sAug[NI][LD2];  // [Quu | I] Gauss-Jordan workspace
  __shared__ float sKt[NI][LD2];   // feedback gain K_t (already negated)
  __shared__ float sx[NA];         // tau_t = [x_t; u_t]
  __shared__ float sg[TT][NS];     // g_t = (C_t tau_t)[:32]
  __shared__ float smu[NS];
  __shared__ float sPivInv;

  const int tid = threadIdx.x;
  const int w   = tid >> 5;  // wave id, 8 waves

  // ---- stage [A|B] into LDS, zero terminal P ----
  for (int e = tid; e < NS * NS; e += 256) sAB[e / NS][e % NS] = Ag[e];
  for (int e = tid; e < NS * NI; e += 256) sAB[e / NI][NS + e % NI] = Bg[e];
  for (int e = tid; e < NS * NS; e += 256) sP[e / NS][e % NS] = 0.0f;
  __syncthreads();

  // ================= backward Riccati sweep =================
  for (int t = TT - 1; t >= 0; --t) {
    const float* Ct = Cg + (size_t)t * NA * NA;
    for (int e = tid; e < NA * NA; e += 256) sCt[e / NA][e % NA] = Ct[e];
    if (t > 0) {  // warm L2/WGP$ for the next step's cost block
      const float* Cn = Cg + (size_t)(t - 1) * NA * NA;
      for (int e = tid * 32; e < NA * NA; e += 256 * 32)
        __builtin_prefetch(&Cn[e], 0, 0);
    }
    __syncthreads();

    // PAB = P @ AB  (32x48, K=32) -> 2x3 tiles
    if (w < 6) {
      const int tm = w / 3, tn = w % 3;
      v8f acc = {};
      acc = wmma_k(&sP[tm * 16][0], LD2, 1, &sAB[0][tn * 16], LD1, 1, NS, acc);
      store_tile(&sPAB[tm * 16][tn * 16], LD1, acc);
    }
    __syncthreads();

    // G = C_t + AB^T @ PAB  (48x48, K=32) -> 3x3 tiles
    for (int tile = w; tile < 9; tile += 8) {
      const int tm = tile / 3, tn = tile % 3;
      v8f acc = load_tile(&sCt[tm * 16][tn * 16], LD1);
      // A operand = AB^T : element (m,k) = sAB[k][tm*16+m]
      acc = wmma_k(&sAB[0][tm * 16], 1, LD1, &sPAB[0][tn * 16], LD1, 1, NS, acc);
      store_tile(&sG[tm * 16][tn * 16], LD1, acc);
    }
    __syncthreads();

    // Aug = [Quu | I] (Quu = G[32:48,32:48], PD -> no pivoting needed)
    for (int e = tid; e < NI * NS; e += 256) {
      const int r = e >> 5, c = e & 31;
      sAug[r][c] = (c < NI) ? sG[NS + r][NS + c] : ((c - NI == r) ? 1.0f : 0.0f);
    }
    __syncthreads();

    // Gauss-Jordan: Aug -> [I | Quu^{-1}]
    for (int p = 0; p < NI; ++p) {
      if (tid == 0) sPivInv = 1.0f / sAug[p][p];
      __syncthreads();
      if (tid < NS) sAug[p][tid] *= sPivInv;
      __syncthreads();
      const int r0 = tid >> 5, r1 = (tid + 256) >> 5;
      const float f0 = sAug[r0][p];
      const float f1 = sAug[r1][p];
      __syncthreads();
      if (r0 != p) sAug[r0][tid & 31] -= f0 * sAug[p][tid & 31];
      if (r1 != p) sAug[r1][tid & 31] -= f1 * sAug[p][tid & 31];
      __syncthreads();
    }

    // K_t = -(Quu^{-1} @ Qux)  (16x32, K=16) -> 1x2 tiles (negate on store)
    if (w < 2) {
      v8f acc = {};
      acc = wmma_k(&sAug[0][NI], LD2, 1, &sG[NS][w * 16], LD1, 1, NI, acc);
      const int lane = tid & 31, half = lane >> 4, l = lane & 15;
#pragma unroll
      for (int r = 0; r < 8; ++r) sKt[r + 8 * half][w * 16 + l] = -acc[r];
    }
    __syncthreads();

    // P_t = Qxx + Qux^T @ K_t  (32x32, K=16) -> 2x2 tiles
    if (w < 4) {
      const int tm = w >> 1, tn = w & 1;
      v8f acc = load_tile(&sG[tm * 16][tn * 16], LD1);
      // A operand = Qux^T : element (m,k) = sG[32+k][tm*16+m]
      acc = wmma_k(&sG[NS][tm * 16], 1, LD1, &sKt[0][tn * 16], LD2, 1, NI, acc);
      store_tile(&sP[tm * 16][tn * 16], LD2, acc);
    }
    // persist K_t for the forward sweep
    for (int e = tid; e < NI * NS; e += 256)
      ws[(size_t)t * NI * NS + e] = sKt[e >> 5][e & 31];
    __syncthreads();
  }

  // ================= forward rollout =================
  if (tid < NS) sx[tid] = x0g[tid];
  __syncthreads();

  for (int t = 0; t < TT; ++t) {
    if (tid < NI) {  // u_t = K_t x_t
      const float* kt = ws + (size_t)t * NI * NS + (size_t)tid * NS;
      float u = 0.f;
#pragma unroll 8
      for (int j = 0; j < NS; ++j) u += kt[j] * sx[j];
      sx[NS + tid] = u;
    }
    __syncthreads();
    if (tid < NA) out[t * NA + tid] = sx[tid];  // tau_t
    float xn = 0.f;
    if (tid < NS) {
      const float* crow = Cg + (size_t)t * NA * NA + (size_t)tid * NA;
      float g = 0.f;
#pragma unroll 8
      for (int j = 0; j < NA; ++j) g += crow[j] * sx[j];
      sg[t][tid] = g;  // g_t = (C_t tau_t)[:32]
#pragma unroll 8
      for (int j = 0; j < NA; ++j) xn += sAB[tid][j] * sx[j];  // x_{t+1}
    }
    __syncthreads();
    if (tid < NS) sx[tid] = xn;
    __syncthreads();
  }

  // ================= multiplier recursion =================
  if (tid < NS) {
    smu[tid] = sg[TT - 1][tid];
    out[TT * NA + (TT - 1) * NS + tid] = smu[tid];
  }
  __syncthreads();
  for (int t = TT - 2; t >= 1; --t) {
    float m = 0.f;
    if (tid < NS) {
      m = sg[t][tid];
#pragma unroll 8
      for (int j = 0; j < NS; ++j) m += sAB[j][tid] * smu[j];  // + A^T mu_{t+1}
    }
    __syncthreads();
    if (tid < NS) {
      smu[tid] = m;
      out[TT * NA + t * NS + tid] = m;
    }
    __syncthreads();
  }
  if (tid < NS) {  // mu_0 has flipped sign (initial-condition constraint uses +I)
    float m = sg[0][tid];
#pragma unroll 8
    for (int j = 0; j < NS; ++j) m += sAB[j][tid] * smu[j];
    out[TT * NA + tid] = -m;
  }
}

extern "C" void kernel_launch(void* const* d_in, const int* in_sizes, int n_in,
                              void* d_out, int out_size, void* d_ws, size_t ws_size,
                              hipStream_t stream) {
  const float* A  = (const float*)d_in[0];
  const float* B  = (const float*)d_in[1];
  const float* C  = (const float*)d_in[2];
  // d_in[3] is T (scalar int) == 64, baked in as TT
  const float* x0 = (const float*)d_in[4];
  float* out = (float*)d_out;
  float* ws  = (float*)d_ws;  // 64 * 16*32 floats = 128 KB of K_t gains
  lqr_riccati_kernel<<<1, 256, 0, stream>>>(A, B, C, x0, out, ws);
}